// mmdFixT_37151467111092
// MI455X (gfx1250) — compile-verified
//
#include <hip/hip_runtime.h>
#include <hip/hip_bf16.h>

// CDNA5 / gfx1250. wave32. WMMA f32 16x16x4 path.
typedef __attribute__((ext_vector_type(2))) float v2f;
typedef __attribute__((ext_vector_type(8))) float v8f;

#define T_DIM 8192
#define D_DIM 64

// ---------------------------------------------------------------- init
__global__ void mmd_init_kernel(float* __restrict__ accum) {
    if (threadIdx.x < 2) accum[threadIdx.x] = 0.0f;
}

// ---------------------------------------------------------------- row norms + term3
// One thread per row: sq[i] = sum_k x[i][k]^2 ; also accumulate sum(exp(sq)).
__global__ void mmd_rownorm_kernel(const float* __restrict__ x,
                                   float* __restrict__ sqv,
                                   float* __restrict__ accum) {
    const int i = blockIdx.x * blockDim.x + threadIdx.x;   // 0..8191
    const float4* row = (const float4*)(x + (size_t)i * D_DIM);
    float s = 0.0f;
#pragma unroll
    for (int q = 0; q < D_DIM / 4; ++q) {
        float4 v = row[q];
        s += v.x * v.x + v.y * v.y + v.z * v.z + v.w * v.w;
    }
    sqv[i] = s;

    // block-reduce exp(sq) into term3 accumulator
    __shared__ float red[256];
    red[threadIdx.x] = __expf(s);
    __syncthreads();
    for (int off = 128; off > 0; off >>= 1) {
        if (threadIdx.x < off) red[threadIdx.x] += red[threadIdx.x + off];
        __syncthreads();
    }
    if (threadIdx.x == 0) atomicAdd(accum + 1, red[0]);
}

// ---------------------------------------------------------------- Gram tile + fused exp + reduce
// Block = 256 threads = 8 waves; each wave computes one 16x16 tile of X*X^T
// with V_WMMA_F32_16X16X4_F32 (K-loop over D=64), then applies
// exp(sq_i + sq_j - 2*g) and reduces. Block tile: 64 rows x 32 cols.
__global__ void __launch_bounds__(256)
mmd_gram_exp_kernel(const float* __restrict__ x,
                    const float* __restrict__ sqv,
                    float* __restrict__ accum) {
    const int tid  = threadIdx.x;
    const int wave = tid >> 5;          // 0..7
    const int lane = tid & 31;          // 0..31
    const int lr   = lane & 15;         // lane row-in-half: 0..15
    const int hi   = lane >> 4;         // 0 | 1 (K-half select for fp32 A/B frags)

    const int ibase = blockIdx.x * 64 + (wave & 3) * 16;   // row tile (M)
    const int jbase = blockIdx.y * 32 + (wave >> 2) * 16;  // col tile (N)

    // fp32 A-fragment layout (16x4): lanes 0-15 hold K={k0,k0+1} of row M=lane,
    // lanes 16-31 hold K={k0+2,k0+3} of row M=lane-16. B (= X^T tile) mirrors it
    // over rows jbase+lr, so both operands load one b64 per step.
    const float* arow = x + (size_t)(ibase + lr) * D_DIM;
    const float* brow = x + (size_t)(jbase + lr) * D_DIM;

    v8f c = {};
#pragma unroll
    for (int k0 = 0; k0 < D_DIM; k0 += 4) {
        const int ka = k0 + hi * 2;
        v2f a = *(const v2f*)(arow + ka);
        v2f b = *(const v2f*)(brow + ka);
        // D = A x B + C  (16x16 f32 accumulator, 8 VGPRs/lane)
        c = __builtin_amdgcn_wmma_f32_16x16x4_f32(
                /*neg_a=*/false, a, /*neg_b=*/false, b,
                /*c_mod=*/(short)0, c, /*reuse_a=*/false, /*reuse_b=*/false);
    }

    // C/D layout: VGPR v -> (M = v + 8*hi, N = lr). Fuse the MMD epilogue.
    const float sqj = sqv[jbase + lr];
    float partial = 0.0f;
#pragma unroll
    for (int v = 0; v < 8; ++v) {
        const float sqi = sqv[ibase + v + hi * 8];
        const float d2  = sqi + sqj - 2.0f * c[v];
        partial += __expf(d2);
    }

    __shared__ float red[256];
    red[tid] = partial;
    __syncthreads();
    for (int off = 128; off > 0; off >>= 1) {
        if (tid < off) red[tid] += red[tid + off];
        __syncthreads();
    }
    if (tid == 0) atomicAdd(accum + 0, red[0]);
}

// ---------------------------------------------------------------- finalize
__global__ void mmd_finalize_kernel(const float* __restrict__ accum,
                                    float* __restrict__ out) {
    const double t1 = (double)accum[0] / ((double)T_DIM * (double)T_DIM);
    const double t3 = (double)accum[1] / (double)T_DIM;
    out[0] = (float)(t1 + 1.0 + t3);
}

// ---------------------------------------------------------------- launch
extern "C" void kernel_launch(void* const* d_in, const int* in_sizes, int n_in,
                              void* d_out, int out_size, void* d_ws, size_t ws_size,
                              hipStream_t stream) {
    (void)in_sizes; (void)n_in; (void)out_size; (void)ws_size;
    const float* x = (const float*)d_in[0];        // xs[0] is the first T*D block
    float* out     = (float*)d_out;

    float* sqv   = (float*)d_ws;                   // [8192] row norms
    float* accum = sqv + T_DIM;                    // [2] {term1_sum, term3_sum}

    mmd_init_kernel<<<1, 64, 0, stream>>>(accum);
    mmd_rownorm_kernel<<<T_DIM / 256, 256, 0, stream>>>(x, sqv, accum);
    dim3 grid(T_DIM / 64, T_DIM / 32);             // 128 x 256 blocks
    mmd_gram_exp_kernel<<<grid, 256, 0, stream>>>(x, sqv, accum);
    mmd_finalize_kernel<<<1, 1, 0, stream>>>(accum, out);
}